// LightGCN_23003844837666
// MI455X (gfx1250) — compile-verified
//
#include <hip/hip_runtime.h>

// ---------------------------------------------------------------------------
// LightGCN propagation for MI455X (gfx1250).
// SpMM is random-scatter => WMMA inapplicable (AI ~0.25 flop/B). Strategy:
//  - wave32 handles 32 edges/batch; each lane owns 2 of the 64 dims
//    => 256B fully-coalesced gather of x[col], scaled, scattered with
//    hardware GLOBAL_ATOMIC_ADD_F32 (no-return) that resolve in the 192MB L2
//    (both 76.8MB tables are L2-resident).
//  - edge stream (rows/cols/vals) staged DOUBLE-BUFFERED into an interleaved
//    16B/edge LDS slab via gfx1250 async copies (per-lane LDS dest addresses
//    do the AoS interleave for free); inner loop = 1 ds_load_b128 per edge.
//  - NNZ (4M) % 32 == 0 => every batch is full, no tail masking anywhere.
// ---------------------------------------------------------------------------

#define LG_NUM_USERS 200000
#define LG_NUM_ITEMS 100000
#define LG_N_NODES   300000
#define LG_DIM       64
#define LG_NNZ       4000000
#define LG_NTOT      (LG_N_NODES * LG_DIM)   // 19,200,000 floats
#define LG_NU        (LG_NUM_USERS * LG_DIM) // 12,800,000 floats (mult of 4)

#if __has_builtin(__builtin_amdgcn_global_load_async_to_lds_b32) && \
    __has_builtin(__builtin_amdgcn_s_wait_asynccnt)
#define LG_ASYNC 1
#else
#define LG_ASYNC 0
#endif

typedef __attribute__((address_space(1))) int* lg_gptr;
typedef __attribute__((address_space(3))) int* lg_lptr;

__device__ __forceinline__ void lg_atomic_add_f32(float* p, float v) {
#if __has_builtin(__builtin_amdgcn_global_atomic_fadd_f32)
  __builtin_amdgcn_global_atomic_fadd_f32(
      (__attribute__((address_space(1))) float*)p, v);
#else
  (void)__hip_atomic_fetch_add(p, v, __ATOMIC_RELAXED, __HIP_MEMORY_SCOPE_AGENT);
#endif
}

// Stage 32 edges starting at e0 into an interleaved LDS slab:
// slab[4*i + 0] = rows[e0+i], [1] = cols, [2] = vals (as b32), [3] = pad.
__device__ __forceinline__ void lg_stage(const int* rows, const int* cols,
                                         const float* vals, int e0, int lane,
                                         int* slab) {
  const int e = e0 + lane;
#if LG_ASYNC
  __builtin_amdgcn_global_load_async_to_lds_b32(
      (lg_gptr)(unsigned long long)(rows + e), (lg_lptr)(slab + 4 * lane + 0), 0, 0);
  __builtin_amdgcn_global_load_async_to_lds_b32(
      (lg_gptr)(unsigned long long)(cols + e), (lg_lptr)(slab + 4 * lane + 1), 0, 0);
  __builtin_amdgcn_global_load_async_to_lds_b32(
      (lg_gptr)(unsigned long long)(vals + e), (lg_lptr)(slab + 4 * lane + 2), 0, 0);
#else
  slab[4 * lane + 0] = rows[e];
  slab[4 * lane + 1] = cols[e];
  slab[4 * lane + 2] = __float_as_int(vals[e]);
#endif
}

// ---------------------------------------------------------------------------
// init: W0 = concat(user,item); acc = same; W1 = 0   (float4 elementwise)
// ---------------------------------------------------------------------------
__global__ __launch_bounds__(256) void lgcn_init(const float* __restrict__ ue,
                                                 const float* __restrict__ ie,
                                                 float* __restrict__ w0,
                                                 float* __restrict__ w1,
                                                 float* __restrict__ acc) {
  const int i = blockIdx.x * blockDim.x + threadIdx.x;  // float4 index
  const int n4 = LG_NTOT / 4;
  if (i >= n4) return;
  float4 v;
  if (i < LG_NU / 4) v = ((const float4*)ue)[i];
  else               v = ((const float4*)ie)[i - LG_NU / 4];
  ((float4*)w0)[i]  = v;
  ((float4*)acc)[i] = v;
  ((float4*)w1)[i]  = make_float4(0.f, 0.f, 0.f, 0.f);
}

// acc += src ; tozero = 0
__global__ __launch_bounds__(256) void lgcn_accum_zero(const float* __restrict__ src,
                                                       float* __restrict__ acc,
                                                       float* __restrict__ tozero) {
  const int i = blockIdx.x * blockDim.x + threadIdx.x;
  const int n4 = LG_NTOT / 4;
  if (i >= n4) return;
  float4 a = ((float4*)acc)[i];
  float4 s = ((const float4*)src)[i];
  a.x += s.x; a.y += s.y; a.z += s.z; a.w += s.w;
  ((float4*)acc)[i]    = a;
  ((float4*)tozero)[i] = make_float4(0.f, 0.f, 0.f, 0.f);
}

// acc = (acc + src) * 0.25
__global__ __launch_bounds__(256) void lgcn_final(const float* __restrict__ src,
                                                  float* __restrict__ acc) {
  const int i = blockIdx.x * blockDim.x + threadIdx.x;
  const int n4 = LG_NTOT / 4;
  if (i >= n4) return;
  float4 a = ((float4*)acc)[i];
  float4 s = ((const float4*)src)[i];
  a.x = (a.x + s.x) * 0.25f; a.y = (a.y + s.y) * 0.25f;
  a.z = (a.z + s.z) * 0.25f; a.w = (a.w + s.w) * 0.25f;
  ((float4*)acc)[i] = a;
}

// ---------------------------------------------------------------------------
// spmm: y[row] += val * x[col], edge-parallel, double-buffered async staging.
// ---------------------------------------------------------------------------
__global__ __launch_bounds__(256) void lgcn_spmm(const float* __restrict__ x,
                                                 const float* __restrict__ vals,
                                                 const int* __restrict__ rows,
                                                 const int* __restrict__ cols,
                                                 float* __restrict__ y) {
  // [buf][wave][edge*4] : 16B-per-edge interleaved records, double buffered
  __shared__ alignas(16) int sE[2][8][32 * 4];

  const int lane   = threadIdx.x & 31;
  const int w      = threadIdx.x >> 5;
  const int gw     = blockIdx.x * 8 + w;
  const int stride = gridDim.x * 8 * 32;   // edges per sweep

  int base = gw * 32;
  if (base >= LG_NNZ) return;              // wave-uniform

  int buf = 0;
  lg_stage(rows, cols, vals, base, lane, &sE[0][w][0]);

  const float* xl = x + 2 * lane;
  float*       yl = y + 2 * lane;

  while (true) {
    const int  nxt  = base + stride;
    const bool more = (nxt < LG_NNZ);      // wave-uniform

#if LG_ASYNC
    if (more) {
      // recycled buffer: make sure its previous ds reads completed
      asm volatile("s_wait_dscnt 0" ::: "memory");
      lg_stage(rows, cols, vals, nxt, lane, &sE[buf ^ 1][w][0]);
      __builtin_amdgcn_s_wait_asynccnt(3); // current batch's 3 loads done
    } else {
      __builtin_amdgcn_s_wait_asynccnt(0);
    }
    asm volatile("" ::: "memory");
#else
    if (more) lg_stage(rows, cols, vals, nxt, lane, &sE[buf ^ 1][w][0]);
    asm volatile("" ::: "memory");
#endif

    // warm L2 for the batch after next (global_prefetch_b8)
    const int pf = base + 2 * stride;
    if (lane == 0 && pf < LG_NNZ) {
      __builtin_prefetch(rows + pf, 0, 3);
      __builtin_prefetch(cols + pf, 0, 3);
      __builtin_prefetch(vals + pf, 0, 3);
    }

    const int* slab = &sE[buf][w][0];
    for (int i = 0; i < 32; ++i) {
      const int4  rcv = *(const int4*)(slab + 4 * i);   // 1 ds_load_b128/edge
      const float vv  = __int_as_float(rcv.z);
      const float2 xv = *(const float2*)(xl + (size_t)rcv.y * LG_DIM); // 256B/wave
      float* yp = yl + (size_t)rcv.x * LG_DIM;
      lg_atomic_add_f32(yp,     vv * xv.x);  // global_atomic_add_f32 no-rtn (L2)
      lg_atomic_add_f32(yp + 1, vv * xv.y);
    }

    if (!more) break;
    base = nxt;
    buf ^= 1;
  }
}

// ---------------------------------------------------------------------------
extern "C" void kernel_launch(void* const* d_in, const int* in_sizes, int n_in,
                              void* d_out, int out_size, void* d_ws, size_t ws_size,
                              hipStream_t stream) {
  const float* ue   = (const float*)d_in[0];  // user_emb  (200000 x 64)
  const float* ie   = (const float*)d_in[1];  // item_emb  (100000 x 64)
  const float* vals = (const float*)d_in[2];  // (4M,)
  const int*   rows = (const int*)d_in[3];    // (4M,)
  const int*   cols = (const int*)d_in[4];    // (4M,)

  float* acc = (float*)d_out;                 // 300000 x 64 = light_out
  float* w0  = (float*)d_ws;                  // ping
  float* w1  = w0 + LG_NTOT;                  // pong (needs 2*76.8MB scratch)

  const int n4 = LG_NTOT / 4;
  dim3 blk(256);
  dim3 grd_ew((n4 + 255) / 256);
  dim3 grd_sp(4096);                          // 32768 persistent waves

  lgcn_init<<<grd_ew, blk, 0, stream>>>(ue, ie, w0, w1, acc);

  // layer 1: emb1 = A @ W0 -> W1 ; acc += W1 ; zero W0
  lgcn_spmm<<<grd_sp, blk, 0, stream>>>(w0, vals, rows, cols, w1);
  lgcn_accum_zero<<<grd_ew, blk, 0, stream>>>(w1, acc, w0);
  // layer 2: emb2 = A @ W1 -> W0 ; acc += W0 ; zero W1
  lgcn_spmm<<<grd_sp, blk, 0, stream>>>(w1, vals, rows, cols, w0);
  lgcn_accum_zero<<<grd_ew, blk, 0, stream>>>(w0, acc, w1);
  // layer 3: emb3 = A @ W0 -> W1 ; acc = (acc + W1) / 4
  lgcn_spmm<<<grd_sp, blk, 0, stream>>>(w0, vals, rows, cols, w1);
  lgcn_final<<<grd_ew, blk, 0, stream>>>(w1, acc);
}